// DecoderLayer_87479893885683
// MI455X (gfx1250) — compile-verified
//
#include <hip/hip_runtime.h>
#include <cstddef>

// ---------------------------------------------------------------------------
// Decoder layer for MI455X (gfx1250), wave32, WMMA f32 16x16x4.
// Memory-bound (~1 GB traffic vs ~1 TFLOP) -> fp32 precision, flash-attention
// streaming, P(X Wv + bv) = (P X) Wv + bv restructure, and double-buffered
// LDS staging via GLOBAL_LOAD_ASYNC_TO_LDS (ASYNCcnt) to overlap HBM/L2
// latency with WMMA in every GEMM.
//
// Workspace layout (floats), SZ = 8192*1024:
//   [0*SZ) q   [1*SZ) k   [2*SZ) Z   [3*SZ) acc   [4*SZ) x
//   [5*SZ) bvmean(1024) colmean(4096)
//   FFN hidden (8192x4096) reuses [0..4*SZ) after attention. ~160 MB total.
// ---------------------------------------------------------------------------

#define B_   4
#define S_   2048
#define D_   1024
#define H_   16
#define KD_  64
#define FF_  4096
#define ROWS_ (B_ * S_)   // 8192

typedef float v2f __attribute__((ext_vector_type(2)));
typedef float v8f __attribute__((ext_vector_type(8)));
typedef int   v4i __attribute__((ext_vector_type(4)));

__device__ __forceinline__ v8f wmma4(v2f a, v2f b, v8f c) {
  // V_WMMA_F32_16X16X4_F32 : D = A(16x4) * B(4x16) + C(16x16), all f32
  return __builtin_amdgcn_wmma_f32_16x16x4_f32(
      /*neg_a=*/false, a, /*neg_b=*/false, b,
      /*c_mod=*/(short)0, c, /*reuse_a=*/false, /*reuse_b=*/false);
}

// ---- async global->LDS copy (gfx1250 GLOBAL_LOAD_ASYNC_TO_LDS_B128) -------
#if __has_builtin(__builtin_amdgcn_global_load_async_to_lds_b128) && \
    __has_builtin(__builtin_amdgcn_s_wait_asynccnt)
#define USE_ASYNC 1
#else
#define USE_ASYNC 0
#endif

__device__ __forceinline__ void cp16_to_lds(const float* g, float* l) {
#if USE_ASYNC
  __builtin_amdgcn_global_load_async_to_lds_b128(
      (__attribute__((address_space(1))) v4i*)g,
      (__attribute__((address_space(3))) v4i*)l, 0, 0);
#else
  const float4 v = *(const float4*)g;
  *(float4*)l = v;
#endif
}

__device__ __forceinline__ void lds_cp_wait() {
#if USE_ASYNC
  __builtin_amdgcn_s_wait_asynccnt(0);
#endif
}

// ---------------------------------------------------------------------------
// Generic row-major GEMM:  C = act(alpha * A[MxK] * B[KxN] + bias)  (or +=)
// Block tile 64x128, BK=16, 8 waves (2x4), 32x32 per wave (2x2 WMMA tiles).
// Double-buffered LDS; next K-tile streams in asynchronously during compute.
// LDS row strides (20, 132 floats) are 16B-aligned and bank-conflict-free.
// ---------------------------------------------------------------------------
template <bool RELU, bool ACCUM>
__global__ __launch_bounds__(256) void gemm_rm(
    const float* __restrict__ A, const float* __restrict__ Bm,
    const float* __restrict__ bias, float* __restrict__ C,
    int M, int N, int K, float alpha) {
  __shared__ float As[2][64][20];
  __shared__ float Bs[2][16][132];
  const int tid  = threadIdx.x;
  const int lane = tid & 31, w = tid >> 5;
  const int hi = lane >> 4, lo = lane & 15;
  const int wm = w >> 2, wn = w & 3;
  const int row0 = blockIdx.x * 64, col0 = blockIdx.y * 128;

  v8f c00 = {}, c01 = {}, c10 = {}, c11 = {};

  const int ar = tid >> 2, ac = (tid & 3) * 4;
  const int br = tid >> 4, bc = (tid & 15) * 8;

  const float* gA  = A + (size_t)(row0 + ar) * K + ac;
  const float* gB0 = Bm + (size_t)br * N + col0 + bc;

  // stage K-tile kt into buffer buf (one b128 for A, two for B, per thread)
#define STAGE(buf, kt)                                                  \
  do {                                                                  \
    cp16_to_lds(gA + (kt), &As[buf][ar][ac]);                           \
    cp16_to_lds(gB0 + (size_t)(kt) * N,     &Bs[buf][br][bc]);          \
    cp16_to_lds(gB0 + (size_t)(kt) * N + 4, &Bs[buf][br][bc + 4]);      \
  } while (0)

  STAGE(0, 0);
  lds_cp_wait();
  __syncthreads();

  int cur = 0;
  for (int kt = 0; kt < K; kt += 16) {
    if (kt + 16 < K) STAGE(cur ^ 1, kt + 16);
#pragma unroll
    for (int kk = 0; kk < 4; ++kk) {
      const int kb = kk * 4 + hi * 2;
      v2f a0, a1, b0, b1;
      a0[0] = As[cur][wm * 32 + lo][kb];      a0[1] = As[cur][wm * 32 + lo][kb + 1];
      a1[0] = As[cur][wm * 32 + 16 + lo][kb]; a1[1] = As[cur][wm * 32 + 16 + lo][kb + 1];
      b0[0] = Bs[cur][kb][wn * 32 + lo];      b0[1] = Bs[cur][kb + 1][wn * 32 + lo];
      b1[0] = Bs[cur][kb][wn * 32 + 16 + lo]; b1[1] = Bs[cur][kb + 1][wn * 32 + 16 + lo];
      c00 = wmma4(a0, b0, c00); c01 = wmma4(a0, b1, c01);
      c10 = wmma4(a1, b0, c10); c11 = wmma4(a1, b1, c11);
    }
    lds_cp_wait();     // next buffer's async copies have landed (this wave)
    __syncthreads();   // all waves done reading cur / writing cur^1
    cur ^= 1;
  }
#undef STAGE
#pragma unroll
  for (int mt = 0; mt < 2; ++mt)
#pragma unroll
    for (int nt = 0; nt < 2; ++nt) {
      v8f cc = (mt == 0) ? (nt == 0 ? c00 : c01) : (nt == 0 ? c10 : c11);
#pragma unroll
      for (int r = 0; r < 8; ++r) {
        const int row = row0 + wm * 32 + mt * 16 + r + hi * 8;
        const int col = col0 + wn * 32 + nt * 16 + lo;
        float v = cc[r] * alpha + (bias ? bias[col] : 0.f);
        if (RELU) v = fmaxf(v, 0.f);
        const size_t idx = (size_t)row * N + col;
        if (ACCUM) C[idx] += v; else C[idx] = v;
      }
    }
}

// ---------------------------------------------------------------------------
// Q/K projection: out[row, h*64+kd] = sum_d in[row,d] * W[h,d,kd] + b[h*64+kd]
// W is (H, D, KD). Each thread's 8-column B segment (col0+bc .. +7, multiple
// of 8) never straddles a 64-wide head block, so the gather is two contiguous
// 16B-aligned b128 async copies: W + (hh*D + kt + br)*KD + kd0 (+4).
// Same async double-buffered pipeline as gemm_rm.
// ---------------------------------------------------------------------------
__global__ __launch_bounds__(256) void qkproj(
    const float* __restrict__ A, const float* __restrict__ W,
    const float* __restrict__ bias, float* __restrict__ C) {
  __shared__ float As[2][64][20];
  __shared__ float Bs[2][16][132];
  const int tid  = threadIdx.x;
  const int lane = tid & 31, w = tid >> 5;
  const int hi = lane >> 4, lo = lane & 15;
  const int wm = w >> 2, wn = w & 3;
  const int row0 = blockIdx.x * 64, col0 = blockIdx.y * 128;

  v8f c00 = {}, c01 = {}, c10 = {}, c11 = {};

  const int ar = tid >> 2, ac = (tid & 3) * 4;
  const int br = tid >> 4, bc = (tid & 15) * 8;

  const int hh  = (col0 + bc) >> 6;     // head of this thread's B segment
  const int kd0 = (col0 + bc) & 63;     // kd offset within head (multiple of 8)

  const float* gA = A + (size_t)(row0 + ar) * D_ + ac;
  const float* gB = W + (size_t)hh * D_ * KD_ + (size_t)br * KD_ + kd0;

#define STAGEQ(buf, kt)                                                   \
  do {                                                                    \
    cp16_to_lds(gA + (kt), &As[buf][ar][ac]);                             \
    cp16_to_lds(gB + (size_t)(kt) * KD_,     &Bs[buf][br][bc]);           \
    cp16_to_lds(gB + (size_t)(kt) * KD_ + 4, &Bs[buf][br][bc + 4]);       \
  } while (0)

  STAGEQ(0, 0);
  lds_cp_wait();
  __syncthreads();

  int cur = 0;
  for (int kt = 0; kt < D_; kt += 16) {
    if (kt + 16 < D_) STAGEQ(cur ^ 1, kt + 16);
#pragma unroll
    for (int kk = 0; kk < 4; ++kk) {
      const int kb = kk * 4 + hi * 2;
      v2f a0, a1, b0, b1;
      a0[0] = As[cur][wm * 32 + lo][kb];      a0[1] = As[cur][wm * 32 + lo][kb + 1];
      a1[0] = As[cur][wm * 32 + 16 + lo][kb]; a1[1] = As[cur][wm * 32 + 16 + lo][kb + 1];
      b0[0] = Bs[cur][kb][wn * 32 + lo];      b0[1] = Bs[cur][kb + 1][wn * 32 + lo];
      b1[0] = Bs[cur][kb][wn * 32 + 16 + lo]; b1[1] = Bs[cur][kb + 1][wn * 32 + 16 + lo];
      c00 = wmma4(a0, b0, c00); c01 = wmma4(a0, b1, c01);
      c10 = wmma4(a1, b0, c10); c11 = wmma4(a1, b1, c11);
    }
    lds_cp_wait();
    __syncthreads();
    cur ^= 1;
  }
#undef STAGEQ
#pragma unroll
  for (int mt = 0; mt < 2; ++mt)
#pragma unroll
    for (int nt = 0; nt < 2; ++nt) {
      v8f cc = (mt == 0) ? (nt == 0 ? c00 : c01) : (nt == 0 ? c10 : c11);
#pragma unroll
      for (int r = 0; r < 8; ++r) {
        const int row = row0 + wm * 32 + mt * 16 + r + hi * 8;
        const int col = col0 + wn * 32 + nt * 16 + lo;
        C[(size_t)row * D_ + col] = cc[r] + bias[col];
      }
    }
}

// ---------------------------------------------------------------------------
// Flash attention for one head: Z[b,s,:] = softmax_row(mask(q k^T / 8)) * X
// Block = 16 query rows, 8 waves each own a 128-wide slice of the 1024 output
// cols. Streams kv blocks of 16 over the causal range; next k-tile is
// prefetched into the alternate LDS buffer via async copies during compute.
// Fully-padded query rows get Z = colmean(X) (all_neg -> uniform softmax).
// ---------------------------------------------------------------------------
__global__ __launch_bounds__(256) void flash_attn(
    const float* __restrict__ Q, const float* __restrict__ Kb,
    const float* __restrict__ X, const unsigned char* __restrict__ pad,
    const float* __restrict__ colmean, float* __restrict__ Z, int h) {
  __shared__ float qs[16][68];
  __shared__ float ks[2][16][68];
  __shared__ float sc[16][17];
  __shared__ float m_sm[16], l_sm[16], a_sm[16], inv_sm[16];
  __shared__ int   padS[16];

  const int tid  = threadIdx.x;
  const int lane = tid & 31, w = tid >> 5;
  const int hi = lane >> 4, lo = lane & 15;
  const int q0 = blockIdx.x * 16;
  const int b  = blockIdx.y;
  const int cw = w * 128;

  const int lr = tid >> 4, lc = (tid & 15) * 4;  // 16x64 tile loader mapping

  // Q tile [16 x 64] + first K tile, asynchronously
  cp16_to_lds(Q + ((size_t)b * S_ + q0 + lr) * D_ + h * KD_ + lc, &qs[lr][lc]);
  cp16_to_lds(Kb + ((size_t)b * S_ + lr) * D_ + h * KD_ + lc, &ks[0][lr][lc]);
  if (tid < 16) {
    m_sm[tid] = -1e30f; l_sm[tid] = 0.f;
    padS[tid] = pad[(size_t)b * S_ + q0 + tid];
  }
  v8f o[8];
#pragma unroll
  for (int nt = 0; nt < 8; ++nt)
#pragma unroll
    for (int r = 0; r < 8; ++r) o[nt][r] = 0.f;

  lds_cp_wait();
  __syncthreads();

  const int nblk = (q0 >> 4) + 1;  // causal range
  int cur = 0;
  for (int tb = 0; tb < nblk; ++tb) {
    const int t0 = tb * 16;
    if (tb + 1 < nblk)  // prefetch next k tile into the other buffer
      cp16_to_lds(Kb + ((size_t)b * S_ + t0 + 16 + lr) * D_ + h * KD_ + lc,
                  &ks[cur ^ 1][lr][lc]);
    if (w == 0) {  // S_tile = q * k^T (16x16, K=64 -> 16 k-steps)
      v8f s8 = {};
#pragma unroll
      for (int ksr = 0; ksr < 16; ++ksr) {
        const int kb = ksr * 4 + hi * 2;
        v2f a, bb;
        a[0]  = qs[lo][kb];          a[1]  = qs[lo][kb + 1];
        bb[0] = ks[cur][lo][kb];     bb[1] = ks[cur][lo][kb + 1];
        s8 = wmma4(a, bb, s8);
      }
#pragma unroll
      for (int r = 0; r < 8; ++r) sc[r + hi * 8][lo] = s8[r] * 0.125f;  // /sqrt(64)
    }
    __syncthreads();
    if (tid < 16) {  // masked online softmax, one row per lane of wave0
      const int row = tid, sg = q0 + row;
      const bool ps = padS[row] != 0;
      const float mold = m_sm[row];
      float mnew = mold;
      float p[16];
#pragma unroll
      for (int j = 0; j < 16; ++j) {
        const int t = t0 + j;
        float sv = -1e30f;
        if (!ps && t <= sg && !pad[(size_t)b * S_ + t]) sv = sc[row][j];
        p[j] = sv;
        mnew = fmaxf(mnew, sv);
      }
      float alpha = 1.f, lsum = 0.f;
      if (mnew > -1e30f) {
        alpha = __expf(mold - mnew);
#pragma unroll
        for (int j = 0; j < 16; ++j) { const float e = __expf(p[j] - mnew); p[j] = e; lsum += e; }
      } else {
#pragma unroll
        for (int j = 0; j < 16; ++j) p[j] = 0.f;
      }
      m_sm[row] = mnew;
      l_sm[row] = l_sm[row] * alpha + lsum;
      a_sm[row] = alpha;
#pragma unroll
      for (int j = 0; j < 16; ++j) sc[row][j] = p[j];
    }
    __syncthreads();
    // rescale running output, then O += P(16x16) * X(16 x 128-slice)
    float ar8[8];
#pragma unroll
    for (int r = 0; r < 8; ++r) ar8[r] = a_sm[r + hi * 8];
#pragma unroll
    for (int nt = 0; nt < 8; ++nt)
#pragma unroll
      for (int r = 0; r < 8; ++r) o[nt][r] *= ar8[r];
#pragma unroll
    for (int ksr = 0; ksr < 4; ++ksr) {
      const int kb = ksr * 4 + hi * 2;
      v2f a;
      a[0] = sc[lo][kb]; a[1] = sc[lo][kb + 1];
      const float* x0 = X + ((size_t)b * S_ + t0 + kb) * D_ + cw;
      const float* x1 = x0 + D_;
#pragma unroll
      for (int nt = 0; nt < 8; ++nt) {
        v2f bb;
        bb[0] = x0[nt * 16 + lo];
        bb[1] = x1[nt * 16 + lo];
        o[nt] = wmma4(a, bb, o[nt]);
      }
    }
    lds_cp_wait();     // prefetched k tile has landed (this wave)
    __syncthreads();   // all waves done with sc / ks[cur]
    cur ^= 1;
  }
  if (tid < 16) inv_sm[tid] = (l_sm[tid] > 0.f) ? 1.f / l_sm[tid] : -1.f;
  __syncthreads();
#pragma unroll
  for (int nt = 0; nt < 8; ++nt)
#pragma unroll
    for (int r = 0; r < 8; ++r) {
      const int row = r + hi * 8;
      const int col = cw + nt * 16 + lo;
      const float iv = inv_sm[row];
      const float val = (iv < 0.f) ? colmean[(size_t)b * D_ + col] : o[nt][r] * iv;
      Z[((size_t)b * S_ + q0 + row) * D_ + col] = val;
    }
}

// ---------------------------------------------------------------------------
// out[row,:] = LayerNorm(a[row,:] + b2[row,:] (+ c3[:])) * w + bias
// ---------------------------------------------------------------------------
__global__ __launch_bounds__(256) void add_ln(
    const float* __restrict__ a, const float* __restrict__ b2,
    const float* __restrict__ c3, const float* __restrict__ w,
    const float* __restrict__ bias, float* __restrict__ out) {
  __shared__ float red[256];
  __shared__ float stat[2];
  const int tid = threadIdx.x;
  const size_t base = (size_t)blockIdx.x * D_;
  float v[4];
  float s = 0.f;
#pragma unroll
  for (int i = 0; i < 4; ++i) {
    const int col = tid + i * 256;
    float t = a[base + col] + b2[base + col];
    if (c3) t += c3[col];
    v[i] = t; s += t;
  }
  red[tid] = s; __syncthreads();
  for (int st = 128; st > 0; st >>= 1) { if (tid < st) red[tid] += red[tid + st]; __syncthreads(); }
  if (tid == 0) stat[0] = red[0] * (1.f / D_);
  __syncthreads();
  const float mu = stat[0];
  s = 0.f;
#pragma unroll
  for (int i = 0; i < 4; ++i) { const float d = v[i] - mu; s += d * d; }
  red[tid] = s; __syncthreads();
  for (int st = 128; st > 0; st >>= 1) { if (tid < st) red[tid] += red[tid + st]; __syncthreads(); }
  if (tid == 0) stat[1] = rsqrtf(red[0] * (1.f / D_) + 1e-5f);
  __syncthreads();
  const float rs = stat[1];
#pragma unroll
  for (int i = 0; i < 4; ++i) {
    const int col = tid + i * 256;
    out[base + col] = (v[i] - mu) * rs * w[col] + bias[col];
  }
}

// colmean[b,e] = mean_s X[b,s,e]   (grid: B * 8 s-chunks * 4 col-chunks)
__global__ __launch_bounds__(256) void colmean_k(const float* __restrict__ X,
                                                 float* __restrict__ cm) {
  const int bid = blockIdx.x;
  const int cchunk = bid & 3, sch = (bid >> 2) & 7, b = bid >> 5;
  const int col = cchunk * 256 + threadIdx.x;
  const int s0 = sch * 256;
  float s = 0.f;
  for (int i = 0; i < 256; ++i)
    s += X[((size_t)b * S_ + s0 + i) * D_ + col];
  atomicAdd(&cm[(size_t)b * D_ + col], s * (1.f / S_));
}

// bvmean[e] = mean_h bv[h,e]
__global__ __launch_bounds__(256) void bvmean_k(const float* __restrict__ bv,
                                                float* __restrict__ o) {
#pragma unroll
  for (int i = 0; i < 4; ++i) {
    const int col = threadIdx.x + i * 256;
    float s = 0.f;
    for (int hh = 0; hh < H_; ++hh) s += bv[(size_t)hh * D_ + col];
    o[col] = s * (1.f / H_);
  }
}

// ---------------------------------------------------------------------------
extern "C" void kernel_launch(void* const* d_in, const int* in_sizes, int n_in,
                              void* d_out, int out_size, void* d_ws, size_t ws_size,
                              hipStream_t stream) {
  const float* input = (const float*)d_in[0];
  const unsigned char* pad = (const unsigned char*)d_in[1];  // jnp bool -> 1 byte
  const float* Wq  = (const float*)d_in[2];
  const float* bq  = (const float*)d_in[3];
  const float* Wk  = (const float*)d_in[4];
  const float* bk  = (const float*)d_in[5];
  const float* Wv  = (const float*)d_in[6];
  const float* bv  = (const float*)d_in[7];
  const float* ln1w = (const float*)d_in[8];
  const float* ln1b = (const float*)d_in[9];
  const float* ln2w = (const float*)d_in[10];
  const float* ln2b = (const float*)d_in[11];
  const float* ff1w = (const float*)d_in[12];
  const float* ff1b = (const float*)d_in[13];
  const float* ff2w = (const float*)d_in[14];
  const float* ff2b = (const float*)d_in[15];

  float* ws = (float*)d_ws;
  const size_t SZ = (size_t)ROWS_ * D_;   // 8M floats
  float* q    = ws + 0 * SZ;
  float* kbuf = ws + 1 * SZ;
  float* Zb   = ws + 2 * SZ;
  float* acc  = ws + 3 * SZ;
  float* x    = ws + 4 * SZ;
  float* hidden = ws;                      // reuses q/k/Z/acc (128 MB) post-attention
  float* bvm  = ws + 5 * SZ;
  float* cm   = bvm + D_;
  float* out  = (float*)d_out;

  (void)hipMemsetAsync(acc, 0, SZ * sizeof(float), stream);
  (void)hipMemsetAsync(cm, 0, (size_t)B_ * D_ * sizeof(float), stream);

  bvmean_k<<<1, 256, 0, stream>>>(bv, bvm);
  colmean_k<<<128, 256, 0, stream>>>(input, cm);

  qkproj<<<dim3(ROWS_ / 64, D_ / 128), 256, 0, stream>>>(input, Wq, bq, q);
  qkproj<<<dim3(ROWS_ / 64, D_ / 128), 256, 0, stream>>>(input, Wk, bk, kbuf);

  for (int h = 0; h < H_; ++h) {
    flash_attn<<<dim3(S_ / 16, B_), 256, 0, stream>>>(q, kbuf, input, pad, cm, Zb, h);
    // acc += (1/H) * Z_h @ Wv_h
    gemm_rm<false, true><<<dim3(ROWS_ / 64, D_ / 128), 256, 0, stream>>>(
        Zb, Wv + (size_t)h * D_ * D_, nullptr, acc, ROWS_, D_, D_, 1.f / H_);
  }

  // x = LN(input + acc + mean_h(bv))
  add_ln<<<ROWS_, 256, 0, stream>>>(input, acc, bvm, ln1w, ln1b, x);

  // hidden = relu(x @ ff1_w + ff1_b)
  gemm_rm<true, false><<<dim3(ROWS_ / 64, FF_ / 128), 256, 0, stream>>>(
      x, ff1w, ff1b, hidden, ROWS_, FF_, D_, 1.f);
  // out = hidden @ ff2_w + ff2_b
  gemm_rm<false, false><<<dim3(ROWS_ / 64, D_ / 128), 256, 0, stream>>>(
      hidden, ff2w, ff2b, out, ROWS_, D_, FF_, 1.f);
  // out = LN(x + out)
  add_ln<<<ROWS_, 256, 0, stream>>>(x, out, nullptr, ln2w, ln2b, out);
}